// SimpleNMS_86775519248474
// MI455X (gfx1250) — compile-verified
//
#include <hip/hip_runtime.h>

// ---------------------------------------------------------------------------
// SimpleNMS (radius-5 fused stencil), MI455X / gfx1250, wave32.
//
// Per 256-thread block: 240x22 output tile, 256x32 input tile (halo 8/5).
// Pipeline (all in one pass, zero-pad == -inf-pad since scores >= 0):
//   M  = (S == maxpool3(S))
//   2x: A = dilate3(M); SS = A ? 0 : S; M |= (SS == maxpool3(SS)) & ~A
//   out = M ? S : 0
// Data movement: global -> LDS via GLOBAL_LOAD_ASYNC_TO_LDS_B128 (ASYNCcnt),
// vertical 3-max in registers, horizontal 3-max via 1-column LDS exchange,
// mask dilation bit-packed in a single u32 per thread.
// Roofline: 128 MiB mandatory HBM traffic -> ~5.6 us @ 23.3 TB/s; fusing all
// 7 pooling/mask stages avoids ~7x multi-kernel traffic.
// ---------------------------------------------------------------------------

#define IMG_W 1024
#define IMG_H 1024
#define NROW 32        // register rows per thread (22 out + 2*5 halo)
#define ROWS_OUT 22
#define COLS_OUT 240   // 256 cols - left halo 8 - right halo 8
#define XOFF 8         // left halo (>=5, multiple of 4 for B128 alignment)

typedef int v4i __attribute__((vector_size(16)));
typedef __attribute__((address_space(1))) v4i as1_v4i;   // global int4
typedef __attribute__((address_space(3))) v4i as3_v4i;   // LDS int4

__device__ __forceinline__ void async_load_b128(const void* gsrc, void* ldst) {
#if defined(__has_builtin)
#if __has_builtin(__builtin_amdgcn_global_load_async_to_lds_b128)
  __builtin_amdgcn_global_load_async_to_lds_b128(
      (as1_v4i*)(unsigned long long)gsrc,
      (as3_v4i*)(unsigned)(unsigned long long)ldst, 0, 0);
  return;
#endif
#endif
  {
    unsigned lofs = (unsigned)(unsigned long long)ldst;   // generic LDS ptr low 32b = LDS offset
    unsigned long long ga = (unsigned long long)gsrc;
    asm volatile("global_load_async_to_lds_b128 %0, %1, off"
                 :: "v"(lofs), "v"(ga) : "memory");
  }
}

__device__ __forceinline__ void wait_async0() {
#if defined(__has_builtin)
#if __has_builtin(__builtin_amdgcn_s_wait_asynccnt)
  __builtin_amdgcn_s_wait_asynccnt(0);
  return;
#endif
#endif
  asm volatile("s_wait_asynccnt 0" ::: "memory");
}

__global__ __launch_bounds__(256)
void simple_nms_kernel(const float* __restrict__ in, float* __restrict__ out) {
  __shared__ float    ldsS[NROW * 256];   // 32 KiB: tile staging + hmax exchange
  __shared__ unsigned ldsM[256];          // bitmask exchange row

  const int x  = threadIdx.x;             // column within tile
  const int bx = blockIdx.x, by = blockIdx.y, bz = blockIdx.z;
  const int gx0 = bx * COLS_OUT - XOFF;   // 4-aligned global column origin
  const int gy0 = by * ROWS_OUT - 5;
  const float* base = in + (size_t)bz * (IMG_W * IMG_H);

  // ---- stage 32x256 fp32 tile into LDS with async B128 copies (8 per lane) ----
  #pragma unroll
  for (int it = 0; it < 8; ++it) {
    int idx = it * 256 + x;               // 0..2047 16-byte chunks
    int row = idx >> 6;                   // 0..31
    int c4  = (idx & 63) << 2;            // column 0,4,...,252
    int gxc = gx0 + c4;
    gxc = gxc < 0 ? 0 : (gxc > IMG_W - 4 ? IMG_W - 4 : gxc);  // stays 4-aligned
    int gyc = gy0 + row;
    gyc = gyc < 0 ? 0 : (gyc > IMG_H - 1 ? IMG_H - 1 : gyc);
    async_load_b128(base + (size_t)gyc * IMG_W + gxc, &ldsS[row * 256 + c4]);
  }
  wait_async0();
  __syncthreads();

  // ---- copy my column into registers, zero-filling outside the image ----
  float S[NROW];
  const int gx = gx0 + x;
  const bool xin = (gx >= 0) && (gx < IMG_W);
  #pragma unroll
  for (int r = 0; r < NROW; ++r) {
    int gy = gy0 + r;
    bool v = xin && (gy >= 0) && (gy < IMG_H);
    S[r] = v ? ldsS[r * 256 + x] : 0.0f;
  }

  const int xm = (x == 0)   ? 0   : x - 1;
  const int xp = (x == 255) ? 255 : x + 1;

  // bit r of result = (X[r] == maxpool3(X)[r]); vertical max in regs,
  // horizontal max via one-column LDS exchange.
  auto eqmax = [&](const float (&X)[NROW]) -> unsigned {
    __syncthreads();                       // protect ldsS WAR vs previous stage
    #pragma unroll
    for (int r = 0; r < NROW; ++r) {
      int rm = (r == 0) ? 0 : r - 1;
      int rp = (r == NROW - 1) ? NROW - 1 : r + 1;
      ldsS[r * 256 + x] = fmaxf(fmaxf(X[rm], X[r]), X[rp]);
    }
    __syncthreads();
    unsigned m = 0;
    #pragma unroll
    for (int r = 0; r < NROW; ++r) {
      float h = fmaxf(fmaxf(ldsS[r * 256 + xm], ldsS[r * 256 + x]),
                      ldsS[r * 256 + xp]);
      m |= (h == X[r]) ? (1u << r) : 0u;
    }
    return m;
  };

  // 3x3 boolean dilation of a bit-packed column: shifts vertically,
  // single-u32 LDS exchange horizontally.
  auto dilate = [&](unsigned bits) -> unsigned {
    unsigned vd = bits | (bits << 1) | (bits >> 1);
    __syncthreads();
    ldsM[x] = vd;
    __syncthreads();
    return ldsM[xm] | ldsM[x] | ldsM[xp];
  };

  unsigned M = eqmax(S);

  float B[NROW];
  #pragma unroll
  for (int k = 0; k < 2; ++k) {
    unsigned A = dilate(M);                       // supp_mask
    #pragma unroll
    for (int r = 0; r < NROW; ++r)
      B[r] = ((A >> r) & 1u) ? 0.0f : S[r];       // supp_scores
    unsigned NM = eqmax(B);                       // new_max_mask
    M |= NM & ~A;
  }

  // ---- store the valid 240x22 center ----
  if (x >= XOFF && x < XOFF + COLS_OUT && gx < IMG_W) {
    float* obase = out + (size_t)bz * (IMG_W * IMG_H);
    #pragma unroll
    for (int r = 5; r < 5 + ROWS_OUT; ++r) {
      int gy = gy0 + r;
      if (gy < IMG_H)
        obase[(size_t)gy * IMG_W + gx] = ((M >> r) & 1u) ? S[r] : 0.0f;
    }
  }
}

extern "C" void kernel_launch(void* const* d_in, const int* in_sizes, int n_in,
                              void* d_out, int out_size, void* d_ws, size_t ws_size,
                              hipStream_t stream) {
  (void)n_in; (void)d_ws; (void)ws_size; (void)out_size;
  const float* scores = (const float*)d_in[0];
  float* out = (float*)d_out;
  int batch = in_sizes[0] / (IMG_W * IMG_H);     // 16
  dim3 block(256, 1, 1);
  dim3 grid((IMG_W + COLS_OUT - 1) / COLS_OUT,   // 5
            (IMG_H + ROWS_OUT - 1) / ROWS_OUT,   // 47
            batch);
  hipLaunchKernelGGL(simple_nms_kernel, grid, block, 0, stream, scores, out);
}